// Model_86088324481732
// MI455X (gfx1250) — compile-verified
//
#include <hip/hip_runtime.h>
#include <math.h>

typedef float v2f __attribute__((ext_vector_type(2)));
typedef float v8f __attribute__((ext_vector_type(8)));

#define IMG_S 256
#define SIGMA_F 3.0e-3f
#define EPS_F 1e-12f
#define DIST_F 2.732f

// ---------------------------------------------------------------------------
// Kernel 1: look_at + perspective projection of vertices -> 2D NDC (xy)
// ---------------------------------------------------------------------------
__global__ void project_k(const float* __restrict__ verts,
                          float* __restrict__ xy, int N) {
    int i = blockIdx.x * blockDim.x + threadIdx.x;
    if (i >= N) return;

    // camera constants (ELEV=0, AZIM=90deg), computed in f32 like the reference
    const float e  = 0.0f;
    const float az = 1.5707963267948966f;
    float ce = cosf(e), se = sinf(e), sa = sinf(az), ca = cosf(az);
    float ex = DIST_F * ce * sa;
    float ey = DIST_F * se;
    float ez = -DIST_F * ce * ca;

    // zax = normalize(at - eye) = normalize(-eye)
    float zl = sqrtf(ex * ex + ey * ey + ez * ez) + EPS_F;
    float zx = -ex / zl, zy = -ey / zl, zz = -ez / zl;
    // xax = normalize(cross(up, zax)), up = (0,1,0)
    float xx = 1.0f * zz - 0.0f * zy;
    float xy_ = 0.0f * zx - 0.0f * zz;
    float xz = 0.0f * zy - 1.0f * zx;
    float xl = sqrtf(xx * xx + xy_ * xy_ + xz * xz) + EPS_F;
    xx /= xl; xy_ /= xl; xz /= xl;
    // yax = cross(zax, xax)
    float yx = zy * xz - zz * xy_;
    float yy = zz * xx - zx * xz;
    float yz = zx * xy_ - zy * xx;

    float vx = verts[3 * i + 0] - ex;
    float vy = verts[3 * i + 1] - ey;
    float vz = verts[3 * i + 2] - ez;

    float cx = vx * xx + vy * xy_ + vz * xz;
    float cy = vx * yx + vy * yy + vz * yz;
    float cz = vx * zx + vy * zy + vz * zz;

    float w = tanf(0.5235987755982988f);  // tan(30 deg)
    float z = fmaxf(cz, 0.01f);
    xy[2 * i + 0] = cx / (z * w);
    xy[2 * i + 1] = cy / (z * w);
}

// ---------------------------------------------------------------------------
// Kernel 2: per-face edge coefficients.  Row layout (A-operand of WMMA):
//   E[(face*4 + slot)*4 + {0,1,2,3}] = {A, B, C, 0}
// so that  d_edge(p) = A*px + B*py + C*1 + 0*0  (K=4 dot product).
// Slot 3 is a dummy edge with C=+1e30 (never wins the min).
// ---------------------------------------------------------------------------
__global__ void edges_k(const float* __restrict__ xy,
                        const int* __restrict__ faces,
                        float* __restrict__ E, int F) {
    int f = blockIdx.x * blockDim.x + threadIdx.x;
    if (f >= F) return;

    int i0 = faces[3 * f + 0], i1 = faces[3 * f + 1], i2 = faces[3 * f + 2];
    float x0 = xy[2 * i0], y0 = xy[2 * i0 + 1];
    float x1 = xy[2 * i1], y1 = xy[2 * i1 + 1];
    float x2 = xy[2 * i2], y2 = xy[2 * i2 + 1];

    float area = (x1 - x0) * (y2 - y0) - (y1 - y0) * (x2 - x0);
    float s = (area >= 0.0f) ? 1.0f : -1.0f;

    float ax[3] = {x0, x1, x2}, ay[3] = {y0, y1, y2};
    float bx[3] = {x1, x2, x0}, by[3] = {y1, y2, y0};

    float4* E4 = (float4*)E;
#pragma unroll
    for (int k = 0; k < 3; ++k) {
        float exk = bx[k] - ax[k];
        float eyk = by[k] - ay[k];
        float t = s / (sqrtf(exk * exk + eyk * eyk) + EPS_F);
        float A = -t * eyk;
        float B = t * exk;
        float C = t * (eyk * ax[k] - exk * ay[k]);
        E4[f * 4 + k] = make_float4(A, B, C, 0.0f);
    }
    E4[f * 4 + 3] = make_float4(0.0f, 0.0f, 1.0e30f, 0.0f);  // dummy slot
}

// ---------------------------------------------------------------------------
// Kernel 3: WMMA rasterization + loss.
// One wave per 16-pixel strip; loop over 256 tiles of 4 faces.
//   A (16x4) : 4 faces x 4 edge slots (one float2 load per lane per tile)
//   B (4x16) : [px, py, 1, 0] per pixel column (constant per wave)
//   D (16x16): signed edge distances; per lane: 2 faces x 4 slots
// ---------------------------------------------------------------------------
__global__ void raster_k(const float* __restrict__ E,
                         const float* __restrict__ ref,
                         float* __restrict__ out, int ntiles) {
    const int lane = threadIdx.x & 31;
    const int wave = blockIdx.x * (blockDim.x >> 5) + (threadIdx.x >> 5);

    const int n   = lane & 15;       // pixel column within strip
    const int h   = lane >> 4;       // 0: K0/K1 half, 1: K2/K3 half
    const int p   = wave * 16 + n;   // linear pixel index
    const int row = p >> 8;
    const int col = p & 255;

    float px = ((float)col + 0.5f) * (2.0f / (float)IMG_S) - 1.0f;
    float py = -(((float)row + 0.5f) * (2.0f / (float)IMG_S) - 1.0f);

    // B operand: lanes 0-15 -> (K0,K1)=(px,py); lanes 16-31 -> (K2,K3)=(1,0)
    v2f b;
    b[0] = (h == 0) ? px : 1.0f;
    b[1] = (h == 0) ? py : 0.0f;

    const float inv_sigma = 1.0f / SIGMA_F;
    const float2* E2 = (const float2*)E;
    const int lidx = n * 2 + h;  // per-lane float2 index within a tile

    float acc = 1.0f;  // running prod over faces of (1 - sigmoid(d/sigma))
    for (int t = 0; t < ntiles; ++t) {
        float2 af = E2[(size_t)t * 32 + lidx];
        v2f a; a[0] = af.x; a[1] = af.y;
        v8f c = {};
        c = __builtin_amdgcn_wmma_f32_16x16x4_f32(
            /*neg_a=*/false, a, /*neg_b=*/false, b,
            /*c_mod=*/(short)0, c, /*reuse_a=*/false, /*reuse_b=*/false);

        float d0 = fminf(fminf(c[0], c[1]), fminf(c[2], c[3]));  // face 4t+0/2
        float d1 = fminf(fminf(c[4], c[5]), fminf(c[6], c[7]));  // face 4t+1/3
        // 1 - sigmoid(d/s) = 1/(1+exp(d/s));  exp overflow -> rcp(inf)=0 (ok)
        float q0 = __builtin_amdgcn_rcpf(1.0f + __expf(d0 * inv_sigma));
        float q1 = __builtin_amdgcn_rcpf(1.0f + __expf(d1 * inv_sigma));
        acc *= q0 * q1;
    }

    // combine the two face-halves held by lane n and lane n+16
    acc *= __shfl_xor(acc, 16, 32);

    float sq = 0.0f;
    if (lane < 16) {
        float sil  = 1.0f - acc;
        float diff = sil - ref[p];
        sq = diff * diff;
    }
#pragma unroll
    for (int off = 16; off > 0; off >>= 1)
        sq += __shfl_down(sq, off, 32);
    if (lane == 0) atomicAdd(out, sq);
}

// ---------------------------------------------------------------------------
extern "C" void kernel_launch(void* const* d_in, const int* in_sizes, int n_in,
                              void* d_out, int out_size, void* d_ws, size_t ws_size,
                              hipStream_t stream) {
    (void)n_in; (void)ws_size;
    const float* verts = (const float*)d_in[0];   // [1,N,3]
    const float* ref   = (const float*)d_in[1];   // [1,S,S]
    const int*   faces = (const int*)d_in[2];     // [1,F,3]

    const int N = in_sizes[0] / 3;  // 512
    const int F = in_sizes[2] / 3;  // 1024

    float* xy = (float*)d_ws;  // N*2 floats
    size_t xy_bytes = ((size_t)N * 2 * sizeof(float) + 255) & ~(size_t)255;
    float* E  = (float*)((char*)d_ws + xy_bytes);  // F*4*4 floats

    hipMemsetAsync(d_out, 0, out_size * sizeof(float), stream);

    project_k<<<(N + 255) / 256, 256, 0, stream>>>(verts, xy, N);
    edges_k<<<(F + 255) / 256, 256, 0, stream>>>(xy, faces, E, F);

    const int ntiles = F / 4;                    // 256 tiles of 4 faces
    const int groups = (IMG_S * IMG_S) / 16;     // 4096 pixel strips (waves)
    const int waves_per_block = 8;               // 256 threads
    raster_k<<<groups / waves_per_block, waves_per_block * 32, 0, stream>>>(
        E, ref, (float*)d_out, ntiles);
}